// RPEMultiHeadAttention_2370821948142
// MI455X (gfx1250) — compile-verified
//
#include <hip/hip_runtime.h>
#include <math.h>

// ---------------------------------------------------------------------------
// RPE multi-head attention with top-k sparse softmax, MI455X (gfx1250).
// All GEMM stages use V_WMMA_F32_16X16X4_F32 (fp32 in/out): the problem is
// HBM-bound (67MB sparse output dominates: written as scores, rewritten as
// probs, read once by the AV GEMM => ~200MB ~ 9us at 23.3 TB/s), so fp32
// precision is free and protects the top-k ranking.
// One wave computes a 16x64 output strip (4 accumulators) in the scores/AV
// stages, so each A fragment feeds 4 WMMAs and the sparse matrix is read
// exactly once; v is stored transposed so B fragments load as 64-bit pairs.
// ---------------------------------------------------------------------------

typedef __attribute__((ext_vector_type(2))) float v2f;
typedef __attribute__((ext_vector_type(8))) float v8f;

#define N_PTS   2048
#define DMODEL  256
#define NHEADS  4
#define DHEAD   64
#define KKTOP   1024
#define HID_ELEMS (N_PTS * DMODEL)

__device__ __forceinline__ v8f wmma_f32_k4(v2f a, v2f b, v8f c) {
  // 8 args: (neg_a, A, neg_b, B, c_mod, C, reuse_a, reuse_b)
  return __builtin_amdgcn_wmma_f32_16x16x4_f32(
      false, a, false, b, (short)0, c, false, false);
}

// monotone order-preserving float -> uint key (descending compare == uint >)
__device__ __forceinline__ unsigned fkey(float f) {
  unsigned u = __float_as_uint(f);
  return (u & 0x80000000u) ? ~u : (u | 0x80000000u);
}
__device__ __forceinline__ float key2f(unsigned k) {
  unsigned u = (k & 0x80000000u) ? (k & 0x7FFFFFFFu) : ~k;
  return __uint_as_float(u);
}

// ---------------------------------------------------------------------------
// 1) QKV projection: out[n][j] = sum_d x[n][d]*W[j][d] + b[j]
//    One wave per 16x16 output tile; K-loop of 64 f32 WMMAs (K=4 each).
//    q,k stored head-major [h][n][dh]; v stored TRANSPOSED [h][dh][n] so the
//    AV GEMM's B-matrix fragments are contiguous 8-byte loads.
//    Store epilogue is branch-free: both layouts are base + n*stride.
// ---------------------------------------------------------------------------
__global__ __launch_bounds__(32) void proj_kernel(
    const float* __restrict__ xq, const float* __restrict__ xk,
    const float* __restrict__ xv,
    const float* __restrict__ Wq, const float* __restrict__ bq,
    const float* __restrict__ Wk, const float* __restrict__ bk,
    const float* __restrict__ Wv, const float* __restrict__ bv,
    float* __restrict__ qh, float* __restrict__ kh, float* __restrict__ vt) {
  const int lane = threadIdx.x & 31;
  const int half = lane >> 4;   // 0: lanes 0-15, 1: lanes 16-31
  const int l    = lane & 15;

  const int tilesPerMat = (N_PTS / 16) * (DMODEL / 16);  // 2048
  int bid = blockIdx.x;
  int mat = bid / tilesPerMat;
  int rem = bid % tilesPerMat;
  int row0 = (rem / (DMODEL / 16)) * 16;
  int col0 = (rem % (DMODEL / 16)) * 16;

  const float* x    = (mat == 0) ? xq : (mat == 1) ? xk : xv;
  const float* W    = (mat == 0) ? Wq : (mat == 1) ? Wk : Wv;
  const float* bias = (mat == 0) ? bq : (mat == 1) ? bk : bv;

  const int rowA = row0 + l;  // A: lane = M row (both halves)
  const int colB = col0 + l;  // B: lane = N col (both halves)

  v8f c = {};
  #pragma unroll 4
  for (int k0 = 0; k0 < DMODEL; k0 += 4) {
    const int ka = k0 + 2 * half;  // lanes 0-15: K,K+1 ; lanes 16-31: K+2,K+3
    v2f a, b;
    a.x = x[rowA * DMODEL + ka];
    a.y = x[rowA * DMODEL + ka + 1];
    b.x = W[colB * DMODEL + ka];   // B[k][j] = W[j][k] (contiguous along k)
    b.y = W[colB * DMODEL + ka + 1];
    c = wmma_f32_k4(a, b, c);
  }

  // D layout: VGPR r -> M=row0+r (lanes 0-15) / row0+r+8 (lanes 16-31), N=l
  const int j  = col0 + l;
  const int h  = j >> 6;
  const int dd = j & 63;
  const float bj = bias[j];

  // unified store addressing: element (n, j) lives at base + n*stride
  float* base;
  size_t stride;
  if (mat == 0) {
    base = qh + (size_t)h * N_PTS * DHEAD + dd;  stride = DHEAD;   // [h][n][d]
  } else if (mat == 1) {
    base = kh + (size_t)h * N_PTS * DHEAD + dd;  stride = DHEAD;   // [h][n][d]
  } else {
    base = vt + ((size_t)h * DHEAD + dd) * N_PTS; stride = 1;      // [h][d][n]
  }
  #pragma unroll
  for (int r = 0; r < 8; ++r) {
    const int n = row0 + r + 8 * half;
    base[(size_t)n * stride] = c[r] + bj;
  }
}

// ---------------------------------------------------------------------------
// 2) Rotary embedding, in-place on q and k (head-major workspace).
//    theta_half[h][n][p] = embed_qk[n][h*32+p], repeated x2 along d.
// ---------------------------------------------------------------------------
__global__ __launch_bounds__(256) void rotary_kernel(
    const float* __restrict__ embed, float* __restrict__ qh,
    float* __restrict__ kh) {
  const int total = NHEADS * N_PTS * (DHEAD / 2);
  int idx = blockIdx.x * blockDim.x + threadIdx.x;
  if (idx >= total) return;
  const int p = idx % (DHEAD / 2);
  const int n = (idx / (DHEAD / 2)) % N_PTS;
  const int h = idx / ((DHEAD / 2) * N_PTS);
  const float t  = embed[n * (DMODEL / 2) + h * (DHEAD / 2) + p];
  const float ct = __cosf(t);
  const float st = __sinf(t);
  const size_t base = ((size_t)(h * N_PTS + n)) * DHEAD + 2 * p;
  const float q0 = qh[base], q1 = qh[base + 1];
  qh[base]     = q0 * ct - q1 * st;
  qh[base + 1] = q1 * ct + q0 * st;
  const float k0 = kh[base], k1 = kh[base + 1];
  kh[base]     = k0 * ct - k1 * st;
  kh[base + 1] = k1 * ct + k0 * st;
}

// ---------------------------------------------------------------------------
// 3) scores[h][n][m] = dot(q[h][n], k[h][m]) / 8 via f32 WMMA.
//    One wave computes a 16x64 strip: each A (q) fragment is loaded once and
//    reused by 4 WMMAs (4 accumulators), quartering q-side traffic.
// ---------------------------------------------------------------------------
__global__ __launch_bounds__(32) void scores_kernel(
    const float* __restrict__ qh, const float* __restrict__ kh,
    float* __restrict__ scores) {
  const int lane = threadIdx.x & 31;
  const int half = lane >> 4;
  const int l    = lane & 15;

  const int tilesPerHead = (N_PTS / 16) * (N_PTS / 64);  // 128*32 = 4096
  int bid = blockIdx.x;
  int h   = bid / tilesPerHead;
  int rem = bid % tilesPerHead;
  int row0 = (rem / (N_PTS / 64)) * 16;
  int col0 = (rem % (N_PTS / 64)) * 64;

  const float* qb = qh + (size_t)h * N_PTS * DHEAD;
  const float* kb = kh + (size_t)h * N_PTS * DHEAD;
  const int rowA = row0 + l;

  v8f c0 = {}, c1 = {}, c2 = {}, c3 = {};
  #pragma unroll 4
  for (int k0 = 0; k0 < DHEAD; k0 += 4) {
    const int ka = k0 + 2 * half;
    v2f a;
    a.x = qb[rowA * DHEAD + ka];
    a.y = qb[rowA * DHEAD + ka + 1];
    v2f b0, b1, b2, b3;
    const int cb = col0 + l;
    b0.x = kb[(cb     ) * DHEAD + ka]; b0.y = kb[(cb     ) * DHEAD + ka + 1];
    b1.x = kb[(cb + 16) * DHEAD + ka]; b1.y = kb[(cb + 16) * DHEAD + ka + 1];
    b2.x = kb[(cb + 32) * DHEAD + ka]; b2.y = kb[(cb + 32) * DHEAD + ka + 1];
    b3.x = kb[(cb + 48) * DHEAD + ka]; b3.y = kb[(cb + 48) * DHEAD + ka + 1];
    c0 = wmma_f32_k4(a, b0, c0);
    c1 = wmma_f32_k4(a, b1, c1);
    c2 = wmma_f32_k4(a, b2, c2);
    c3 = wmma_f32_k4(a, b3, c3);
  }

  float* sp = scores + (size_t)h * N_PTS * N_PTS;
  #pragma unroll
  for (int r = 0; r < 8; ++r) {
    const int n = row0 + r + 8 * half;
    float* dst = sp + (size_t)n * N_PTS + col0 + l;
    dst[ 0] = c0[r] * 0.125f;   // 1/sqrt(64)
    dst[16] = c1[r] * 0.125f;
    dst[32] = c2[r] * 0.125f;
    dst[48] = c3[r] * 0.125f;
  }
}

// ---------------------------------------------------------------------------
// 4) Per-row top-1024 + softmax, in place. One 256-thread block per row.
//    Exact threshold via 4-pass 8-bit radix select on the monotone key.
//    All ties share one exact bit pattern, so the softmax sum is exact;
//    tie positions beyond the needed count are zeroed via a shared counter.
// ---------------------------------------------------------------------------
__global__ __launch_bounds__(256) void topk_softmax_kernel(
    float* __restrict__ scores) {
  __shared__ float    row[N_PTS];
  __shared__ unsigned hist[256];
  __shared__ float    red[256];
  __shared__ unsigned s_prefix;
  __shared__ int      s_remaining;
  __shared__ unsigned s_tie;

  const int tid = threadIdx.x;
  const size_t rbase = (size_t)blockIdx.x * N_PTS;  // blockIdx over H*N rows

  // load row + compute row max (== max of selected set, since top-1 selected)
  float mx = -INFINITY;
  for (int m = tid; m < N_PTS; m += 256) {
    const float v = scores[rbase + m];
    row[m] = v;
    mx = fmaxf(mx, v);
  }
  red[tid] = mx;
  __syncthreads();
  for (int s = 128; s > 0; s >>= 1) {
    if (tid < s) red[tid] = fmaxf(red[tid], red[tid + s]);
    __syncthreads();
  }
  const float rowmax = red[0];

  if (tid == 0) { s_prefix = 0u; s_remaining = KKTOP; }
  __syncthreads();

  for (int shift = 24; shift >= 0; shift -= 8) {
    hist[tid] = 0u;
    __syncthreads();
    const unsigned prefix = s_prefix;
    for (int m = tid; m < N_PTS; m += 256) {
      const unsigned u = fkey(row[m]);
      const bool match = (shift == 24) || (((u ^ prefix) >> (shift + 8)) == 0u);
      if (match) atomicAdd(&hist[(u >> shift) & 255u], 1u);
    }
    __syncthreads();
    if (tid == 0) {
      const int rem = s_remaining;
      unsigned cum = 0u;
      for (int bin = 255; bin >= 0; --bin) {
        const unsigned cnt = hist[bin];
        if (cum + cnt >= (unsigned)rem) {
          s_prefix    = prefix | ((unsigned)bin << shift);
          s_remaining = rem - (int)cum;  // ties still needed inside this bin
          break;
        }
        cum += cnt;
      }
    }
    __syncthreads();
  }
  const unsigned T = s_prefix;         // key of the 1024-th largest element
  const int tiesNeeded = s_remaining;  // how many u==T elements to keep
  const float tval = key2f(T);

  // sum of exp over selected (strictly greater + tiesNeeded copies of tval)
  float partial = 0.f;
  for (int m = tid; m < N_PTS; m += 256) {
    if (fkey(row[m]) > T) partial += __expf(row[m] - rowmax);
  }
  red[tid] = partial;
  __syncthreads();
  for (int s = 128; s > 0; s >>= 1) {
    if (tid < s) red[tid] += red[tid + s];
    __syncthreads();
  }
  const float sumexp  = red[0] + (float)tiesNeeded * __expf(tval - rowmax);
  const float inv     = 1.0f / sumexp;
  const float tieprob = __expf(tval - rowmax) * inv;

  if (tid == 0) s_tie = 0u;
  __syncthreads();

  for (int m = tid; m < N_PTS; m += 256) {
    const unsigned u = fkey(row[m]);
    float out = 0.f;
    if (u > T) {
      out = __expf(row[m] - rowmax) * inv;
    } else if (u == T) {
      const unsigned pos = atomicAdd(&s_tie, 1u);
      if (pos < (unsigned)tiesNeeded) out = tieprob;
    }
    scores[rbase + m] = out;
  }
}

// ---------------------------------------------------------------------------
// 5) hidden[n][h*64+d] = sum_m sparse[h][n][m] * v[h][m][d]  (f32 WMMA).
//    One wave per 16-row strip computes ALL 64 output columns (4 accums), so
//    the 67MB sparse matrix is read exactly once. v is pre-transposed
//    ([h][d][m]) so B fragments are contiguous 8-byte loads.
// ---------------------------------------------------------------------------
__global__ __launch_bounds__(32) void av_kernel(
    const float* __restrict__ sparse, const float* __restrict__ vt,
    float* __restrict__ hidden) {
  const int lane = threadIdx.x & 31;
  const int half = lane >> 4;
  const int l    = lane & 15;

  int bid  = blockIdx.x;                 // NHEADS * (N_PTS/16) = 512
  int h    = bid / (N_PTS / 16);
  int row0 = (bid % (N_PTS / 16)) * 16;

  const float* A = sparse + (size_t)h * N_PTS * N_PTS;  // [n][m]
  const float* V = vt     + (size_t)h * DHEAD * N_PTS;  // [d][m] (transposed)
  const int rowA = row0 + l;

  v8f c0 = {}, c1 = {}, c2 = {}, c3 = {};
  #pragma unroll 4
  for (int k0 = 0; k0 < N_PTS; k0 += 4) {
    const int ka = k0 + 2 * half;
    v2f a;
    a.x = A[(size_t)rowA * N_PTS + ka];
    a.y = A[(size_t)rowA * N_PTS + ka + 1];
    v2f b0, b1, b2, b3;
    b0.x = V[(size_t)(l     ) * N_PTS + ka]; b0.y = V[(size_t)(l     ) * N_PTS + ka + 1];
    b1.x = V[(size_t)(l + 16) * N_PTS + ka]; b1.y = V[(size_t)(l + 16) * N_PTS + ka + 1];
    b2.x = V[(size_t)(l + 32) * N_PTS + ka]; b2.y = V[(size_t)(l + 32) * N_PTS + ka + 1];
    b3.x = V[(size_t)(l + 48) * N_PTS + ka]; b3.y = V[(size_t)(l + 48) * N_PTS + ka + 1];
    c0 = wmma_f32_k4(a, b0, c0);
    c1 = wmma_f32_k4(a, b1, c1);
    c2 = wmma_f32_k4(a, b2, c2);
    c3 = wmma_f32_k4(a, b3, c3);
  }

  #pragma unroll
  for (int r = 0; r < 8; ++r) {
    const int n = row0 + r + 8 * half;
    float* dst = hidden + (size_t)n * DMODEL + h * DHEAD + l;
    dst[ 0] = c0[r];
    dst[16] = c1[r];
    dst[32] = c2[r];
    dst[48] = c3[r];
  }
}

// ---------------------------------------------------------------------------
extern "C" void kernel_launch(void* const* d_in, const int* in_sizes, int n_in,
                              void* d_out, int out_size, void* d_ws,
                              size_t ws_size, hipStream_t stream) {
  const float* xq    = (const float*)d_in[0];
  const float* xk    = (const float*)d_in[1];
  const float* xv    = (const float*)d_in[2];
  const float* embed = (const float*)d_in[3];
  const float* Wq    = (const float*)d_in[4];
  const float* bq    = (const float*)d_in[5];
  const float* Wk    = (const float*)d_in[6];
  const float* bk    = (const float*)d_in[7];
  const float* Wv    = (const float*)d_in[8];
  const float* bv    = (const float*)d_in[9];
  (void)in_sizes; (void)n_in; (void)out_size; (void)ws_size;

  float* out    = (float*)d_out;
  float* hidden = out;               // N*DMODEL floats
  float* sparse = out + HID_ELEMS;   // H*N*N floats (scores -> probs in place)

  // workspace: q,k head-major + v transposed, 3 * 4*2048*64 floats = 6 MB
  float* qh = (float*)d_ws;
  float* kh = qh + (size_t)NHEADS * N_PTS * DHEAD;
  float* vt = kh + (size_t)NHEADS * N_PTS * DHEAD;

  proj_kernel<<<3 * (N_PTS / 16) * (DMODEL / 16), 32, 0, stream>>>(
      xq, xk, xv, Wq, bq, Wk, bk, Wv, bv, qh, kh, vt);

  rotary_kernel<<<(NHEADS * N_PTS * (DHEAD / 2) + 255) / 256, 256, 0, stream>>>(
      embed, qh, kh);

  scores_kernel<<<NHEADS * (N_PTS / 16) * (N_PTS / 64), 32, 0, stream>>>(
      qh, kh, sparse);

  topk_softmax_kernel<<<NHEADS * N_PTS, 256, 0, stream>>>(sparse);

  av_kernel<<<NHEADS * (N_PTS / 16), 32, 0, stream>>>(
      sparse, vt, hidden);
}